// QMAMLModelDiff_14817637171323
// MI455X (gfx1250) — compile-verified
//
#include <hip/hip_runtime.h>
#include <math.h>

// ---------------------------------------------------------------------------
// QMAML fused pipeline for MI455X (gfx1250).
//
// Math reduction: the SEL circuit (theta_shared+theta_task) is batch-invariant
// -> precompute fixed 64x64 complex unitary U once (kernel A).  RY-encoded
// |0..0> is a REAL product state v (64 floats / row).  <Z_w> readout + Wc fold
// into real 64x5 matrix G.  Main kernel (B) is then 4 WMMA-f32 GEMM stages +
// elementwise glue, ~20 MB HBM traffic total -> memory-bound at 23.3 TB/s.
// fp32 WMMA (V_WMMA_F32_16X16X4_F32) keeps reference-grade numerics.
// ---------------------------------------------------------------------------

#define NQ 6
#define DIM 64
#define BATCH 65536
#define ROWS_PER_WAVE 16
#define WAVES_PER_BLOCK 8
#define ROWS_PER_BLOCK (ROWS_PER_WAVE * WAVES_PER_BLOCK)   // 128
#define SSTRIDE 65                                         // pad to avoid LDS bank conflicts
#define WAVE_SMEM (16 * SSTRIDE + 96 + 96)                 // scratch + cbuf + sbuf (floats)

typedef __attribute__((ext_vector_type(2))) float v2f;
typedef __attribute__((ext_vector_type(8))) float v8f;

static __device__ __forceinline__ v8f vzero8() {
  v8f z = {0.f, 0.f, 0.f, 0.f, 0.f, 0.f, 0.f, 0.f};
  return z;
}

// D = A(16x4,f32) * B(4x16,f32) + C(16x16,f32); wave32, EXEC all-ones.
static __device__ __forceinline__ v8f wmma4(v2f a, v2f b, v8f c) {
  return __builtin_amdgcn_wmma_f32_16x16x4_f32(false, a, false, b, (short)0, c,
                                               false, false);
}

// ---------------------------------------------------------------------------
// Kernel A: build U = SEL_task(SEL_shared(.)) as a 64x64 complex matrix by
// applying the gate sequence to identity columns in LDS; emit U^T (real/imag)
// into workspace in B-matrix row-major [k=j][n=i] order.
// ---------------------------------------------------------------------------
__global__ __launch_bounds__(64) void build_unitary_kernel(
    const float* __restrict__ theta_shared, const float* __restrict__ theta_task,
    float* __restrict__ UrT, float* __restrict__ UiT) {
  __shared__ float Sr[DIM * DIM];
  __shared__ float Si[DIM * DIM];
  const int tid = threadIdx.x;

  for (int idx = tid; idx < DIM * DIM; idx += 64) {
    Sr[idx] = ((idx >> 6) == (idx & 63)) ? 1.f : 0.f;
    Si[idx] = 0.f;
  }
  __syncthreads();

  for (int part = 0; part < 2; ++part) {
    const float* th = (part == 0) ? theta_shared : theta_task;
    const int L = (part == 0) ? 2 : 1;
    for (int l = 0; l < L; ++l) {
      // Rot(phi,th,om) = RZ(om) RY(th) RZ(phi) on each wire
      for (int w = 0; w < NQ; ++w) {
        const float phi = th[(l * NQ + w) * 3 + 0];
        const float tht = th[(l * NQ + w) * 3 + 1];
        const float om  = th[(l * NQ + w) * 3 + 2];
        float sA, cA, sB, cB, st, ct;
        sincosf(0.5f * (phi + om), &sA, &cA);
        sincosf(0.5f * (phi - om), &sB, &cB);
        sincosf(0.5f * tht, &st, &ct);
        const float u00r =  ct * cA, u00i = -ct * sA;   // ct*e^{-iA}
        const float u01r = -st * cB, u01i = -st * sB;   // -st*e^{+iB}
        const float u10r =  st * cB, u10i = -st * sB;   // st*e^{-iB}
        const float u11r =  ct * cA, u11i =  ct * sA;   // ct*e^{+iA}
        const int pb = 5 - w, bit = 1 << pb;
        for (int task = tid; task < 32 * DIM; task += 64) {
          const int col = task & 63, p = task >> 6;
          const int i0 = ((p >> pb) << (pb + 1)) | (p & (bit - 1));
          const int i1 = i0 | bit;
          const float ar = Sr[i0 * DIM + col], ai = Si[i0 * DIM + col];
          const float br = Sr[i1 * DIM + col], bi = Si[i1 * DIM + col];
          Sr[i0 * DIM + col] = u00r * ar - u00i * ai + u01r * br - u01i * bi;
          Si[i0 * DIM + col] = u00r * ai + u00i * ar + u01r * bi + u01i * br;
          Sr[i1 * DIM + col] = u10r * ar - u10i * ai + u11r * br - u11i * bi;
          Si[i1 * DIM + col] = u10r * ai + u10i * ar + u11r * bi + u11i * br;
        }
        __syncthreads();
      }
      // ring of CNOTs with range r
      const int r = l % (NQ - 1) + 1;
      for (int w = 0; w < NQ; ++w) {
        const int tq = (w + r) % NQ;
        const int pc = 5 - w, pt = 5 - tq;
        const int lo = pc < pt ? pc : pt, hi = pc < pt ? pt : pc;
        for (int task = tid; task < 16 * DIM; task += 64) {
          const int col = task & 63;
          const int p = task >> 6;
          int i = ((p >> lo) << (lo + 1)) | (p & ((1 << lo) - 1));
          i = ((i >> hi) << (hi + 1)) | (i & ((1 << hi) - 1));
          const int ia = i | (1 << pc);        // control=1, target=0
          const int ib = ia | (1 << pt);       // control=1, target=1
          float t0 = Sr[ia * DIM + col];
          Sr[ia * DIM + col] = Sr[ib * DIM + col];
          Sr[ib * DIM + col] = t0;
          t0 = Si[ia * DIM + col];
          Si[ia * DIM + col] = Si[ib * DIM + col];
          Si[ib * DIM + col] = t0;
        }
        __syncthreads();
      }
    }
  }
  for (int idx = tid; idx < DIM * DIM; idx += 64) {
    const int i = idx >> 6, j = idx & 63;
    UrT[j * DIM + i] = Sr[i * DIM + j];
    UiT[j * DIM + i] = Si[i * DIM + j];
  }
}

// ---------------------------------------------------------------------------
// Kernel B: fused encoder + state-vector contraction + readout.
// One wave per 16-row batch tile; 8 waves/block; all GEMMs via WMMA f32.
// ---------------------------------------------------------------------------
__global__ __launch_bounds__(256) void qmaml_fused_kernel(
    const float* __restrict__ x, const float* __restrict__ W1,
    const float* __restrict__ b1, const float* __restrict__ ln_g,
    const float* __restrict__ ln_b, const float* __restrict__ W2,
    const float* __restrict__ b2, const float* __restrict__ Wc,
    const float* __restrict__ bc, const float* __restrict__ UrT,
    const float* __restrict__ UiT, float* __restrict__ out) {
  extern __shared__ float smem[];
  float* sW1T = smem;                 // 4096: W1^T  [d][o]
  float* sUrT = sW1T + 4096;          // 4096: Ur^T  [j][i]
  float* sUiT = sUrT + 4096;          // 4096: Ui^T  [j][i]
  float* sW2T = sUiT + 4096;          // 64*16: W2^T padded [d][n] (n>=6 -> 0)
  float* sG   = sW2T + 1024;          // 64*16: G padded    [i][k] (k>=5 -> 0)
  float* swav = sG + 1024;            // per-wave scratch

  const int tid = threadIdx.x;
  const int lane = tid & 31;
  const int wave = tid >> 5;

  // ---- cooperative weight staging ----
  for (int idx = tid; idx < 4096; idx += 256) {
    const int o = idx >> 6, d = idx & 63;
    sW1T[d * 64 + o] = W1[o * 64 + d];
    sUrT[idx] = UrT[idx];
    sUiT[idx] = UiT[idx];
  }
  for (int idx = tid; idx < 1024; idx += 256) {
    const int d = idx >> 4, n = idx & 15;
    sW2T[idx] = (n < NQ) ? W2[n * 64 + d] : 0.f;
    float g = 0.f;
    if (n < 5) {
#pragma unroll
      for (int w = 0; w < NQ; ++w) {
        const float sgn = ((d >> (5 - w)) & 1) ? -1.f : 1.f;
        g += sgn * Wc[n * NQ + w];
      }
    }
    sG[idx] = g;
  }
  __syncthreads();

  float* scratch = swav + wave * WAVE_SMEM;   // 16 x SSTRIDE tile buffer
  float* cbuf = scratch + 16 * SSTRIDE;       // 16 x 6 cos
  float* sbuf = cbuf + 96;                    // 16 x 6 sin

  const int rowBase = blockIdx.x * ROWS_PER_BLOCK + wave * ROWS_PER_WAVE;

  // WMMA fragment lane mappings (f32 16x16x4, wave32):
  const int M = lane & 15;             // A row
  const int N = lane & 15;             // B/C col within 16-tile
  const int koff = (lane >> 4) * 2;    // A: K offset {0,2}, vgpr adds +0/+1
  const int kgrp = (lane >> 4) * 2;    // B: K row group {0,2}
  const int mBase = (lane >> 4) * 8;   // C: M = r + mBase

  // ---- stage 0: stage x tile into LDS (fully coalesced) ----
  for (int idx = lane; idx < 1024; idx += 32)
    scratch[(idx >> 6) * SSTRIDE + (idx & 63)] = x[rowBase * 64 + idx];
  __syncthreads();

  // ---- stage 1: H = relu(x @ W1^T + b1), 4 N-tiles, K=64 ----
  v8f hc[4];
#pragma unroll
  for (int nt = 0; nt < 4; ++nt) hc[nt] = vzero8();
  for (int kb = 0; kb < 64; kb += 4) {
    v2f a;
    a.x = scratch[M * SSTRIDE + kb + koff];
    a.y = scratch[M * SSTRIDE + kb + koff + 1];
#pragma unroll
    for (int nt = 0; nt < 4; ++nt) {
      v2f b;
      b.x = sW1T[(kb + kgrp) * 64 + nt * 16 + N];
      b.y = sW1T[(kb + kgrp + 1) * 64 + nt * 16 + N];
      hc[nt] = wmma4(a, b, hc[nt]);
    }
  }
  __syncthreads();
#pragma unroll
  for (int nt = 0; nt < 4; ++nt) {
    const int col = nt * 16 + N;
    const float bias = b1[col];
#pragma unroll
    for (int r = 0; r < 8; ++r)
      scratch[(mBase + r) * SSTRIDE + col] = fmaxf(hc[nt][r] + bias, 0.f);
  }
  __syncthreads();

  // ---- stage 2: LayerNorm in-place (2 lanes per row, shfl_xor combine) ----
  {
    const int row = lane & 15;
    const int d0 = (lane >> 4) * 32;
    float ssum = 0.f;
    for (int d = d0; d < d0 + 32; ++d) ssum += scratch[row * SSTRIDE + d];
    ssum += __shfl_xor(ssum, 16, 32);
    const float mu = ssum * (1.f / 64.f);
    float vsum = 0.f;
    for (int d = d0; d < d0 + 32; ++d) {
      const float t = scratch[row * SSTRIDE + d] - mu;
      vsum += t * t;
    }
    vsum += __shfl_xor(vsum, 16, 32);
    const float inv = rsqrtf(vsum * (1.f / 64.f) + 1e-5f);
    for (int d = d0; d < d0 + 32; ++d) {
      const float t = (scratch[row * SSTRIDE + d] - mu) * inv;
      scratch[row * SSTRIDE + d] = t * ln_g[d] + ln_b[d];
    }
  }
  __syncthreads();

  // ---- stage 3: z = tanh(H @ W2^T + b2), then cos/sin(pi*z/2) ----
  v8f zc = vzero8();
  for (int kb = 0; kb < 64; kb += 4) {
    v2f a;
    a.x = scratch[M * SSTRIDE + kb + koff];
    a.y = scratch[M * SSTRIDE + kb + koff + 1];
    v2f b;
    b.x = sW2T[(kb + kgrp) * 16 + N];
    b.y = sW2T[(kb + kgrp + 1) * 16 + N];
    zc = wmma4(a, b, zc);
  }
  if (N < NQ) {
    const float bias = b2[N];
#pragma unroll
    for (int r = 0; r < 8; ++r) {
      const float z = tanhf(zc[r] + bias);
      float s, c;
      sincosf(1.57079632679489662f * z, &s, &c);   // pi*z/2
      cbuf[(mBase + r) * NQ + N] = c;
      sbuf[(mBase + r) * NQ + N] = s;
    }
  }
  __syncthreads();

  // ---- stage 4: real product-state amplitudes v[b][i] ----
  for (int idx = lane; idx < 1024; idx += 32) {
    const int bRow = idx >> 6, i = idx & 63;
    float v = 1.f;
#pragma unroll
    for (int w = 0; w < NQ; ++w)
      v *= ((i >> (5 - w)) & 1) ? sbuf[bRow * NQ + w] : cbuf[bRow * NQ + w];
    scratch[bRow * SSTRIDE + i] = v;
  }
  __syncthreads();

  // ---- stage 5: yr = V @ Ur^T, yi = V @ Ui^T ----
  v8f yr[4], yi[4];
#pragma unroll
  for (int nt = 0; nt < 4; ++nt) { yr[nt] = vzero8(); yi[nt] = vzero8(); }
  for (int kb = 0; kb < 64; kb += 4) {
    v2f a;
    a.x = scratch[M * SSTRIDE + kb + koff];
    a.y = scratch[M * SSTRIDE + kb + koff + 1];
#pragma unroll
    for (int nt = 0; nt < 4; ++nt) {
      v2f br, bi;
      br.x = sUrT[(kb + kgrp) * 64 + nt * 16 + N];
      br.y = sUrT[(kb + kgrp + 1) * 64 + nt * 16 + N];
      bi.x = sUiT[(kb + kgrp) * 64 + nt * 16 + N];
      bi.y = sUiT[(kb + kgrp + 1) * 64 + nt * 16 + N];
      yr[nt] = wmma4(a, br, yr[nt]);
      yi[nt] = wmma4(a, bi, yi[nt]);
    }
  }
  __syncthreads();
  // probabilities p = yr^2 + yi^2 -> scratch
#pragma unroll
  for (int nt = 0; nt < 4; ++nt) {
#pragma unroll
    for (int r = 0; r < 8; ++r) {
      const float pr = yr[nt][r], pi = yi[nt][r];
      scratch[(mBase + r) * SSTRIDE + nt * 16 + N] = pr * pr + pi * pi;
    }
  }
  __syncthreads();

  // ---- stage 6: out = P @ G + bc (5 valid cols) ----
  v8f oc = vzero8();
  for (int kb = 0; kb < 64; kb += 4) {
    v2f a;
    a.x = scratch[M * SSTRIDE + kb + koff];
    a.y = scratch[M * SSTRIDE + kb + koff + 1];
    v2f b;
    b.x = sG[(kb + kgrp) * 16 + N];
    b.y = sG[(kb + kgrp + 1) * 16 + N];
    oc = wmma4(a, b, oc);
  }
  if (N < 5) {
    const float bias = bc[N];
#pragma unroll
    for (int r = 0; r < 8; ++r)
      out[(rowBase + mBase + r) * 5 + N] = oc[r] + bias;
  }
}

// ---------------------------------------------------------------------------
extern "C" void kernel_launch(void* const* d_in, const int* in_sizes, int n_in,
                              void* d_out, int out_size, void* d_ws,
                              size_t ws_size, hipStream_t stream) {
  (void)in_sizes; (void)n_in; (void)out_size; (void)ws_size;
  const float* x   = (const float*)d_in[0];
  const float* W1  = (const float*)d_in[1];
  const float* b1  = (const float*)d_in[2];
  const float* lng = (const float*)d_in[3];
  const float* lnb = (const float*)d_in[4];
  const float* W2  = (const float*)d_in[5];
  const float* b2  = (const float*)d_in[6];
  const float* ths = (const float*)d_in[7];
  const float* tht = (const float*)d_in[8];
  const float* Wc  = (const float*)d_in[9];
  const float* bc  = (const float*)d_in[10];
  float* out = (float*)d_out;
  float* UrT = (float*)d_ws;          // 4096 floats
  float* UiT = UrT + 4096;            // 4096 floats

  build_unitary_kernel<<<1, 64, 0, stream>>>(ths, tht, UrT, UiT);

  const size_t smem =
      (size_t)(3 * 4096 + 2 * 1024 + WAVES_PER_BLOCK * WAVE_SMEM) * sizeof(float);
  qmaml_fused_kernel<<<BATCH / ROWS_PER_BLOCK, 256, smem, stream>>>(
      x, W1, b1, lng, lnb, W2, b2, Wc, bc, UrT, UiT, out);
}